// HausdorffLoss_63445256896852
// MI455X (gfx1250) — compile-verified
//
#include <hip/hip_runtime.h>

typedef __attribute__((ext_vector_type(2))) float v2f;
typedef __attribute__((ext_vector_type(8))) float v8f;

#define NB 8
#define NV 4096
#define NWAVES 8           // 256 threads/block = 8 wave32
#define COLS_PER_WG 1024   // column chunk per workgroup
#define NTILES (COLS_PER_WG / 16)
#define INF_BITS 0x7F800000u

// ds_swizzle_b32 group-of-32 encodings: offset[4:0]=and, [9:5]=or, [14:10]=xor
#define SWZ_XOR1  0x041f
#define SWZ_XOR2  0x081f
#define SWZ_XOR4  0x101f
#define SWZ_XOR8  0x201f
#define SWZ_XOR16 0x401f
#define SWZF(v, m) __int_as_float(__builtin_amdgcn_ds_swizzle(__float_as_int(v), (m)))
#define SWZU(v, m) ((unsigned)__builtin_amdgcn_ds_swizzle((int)(v), (m)))

__device__ __forceinline__ unsigned umin2(unsigned a, unsigned b) { return a < b ? a : b; }

// ---------------------------------------------------------------------------
// Init workspace: colminU[NB][NV] and rowminU[NB][NV] to +inf (float bits).
// ---------------------------------------------------------------------------
__global__ void hausdorff_init(unsigned* __restrict__ p, int count) {
    int i = blockIdx.x * blockDim.x + threadIdx.x;
    if (i < count) p[i] = INF_BITS;
}

// ---------------------------------------------------------------------------
// Main kernel: one WMMA f32 16x16x4 per 16x16 tile of squared distances.
//   A = [-2x0,-2x1 | -2x2, x2[M]] , B = [y0,y1 | y2, 1]
//   => D[M][N] = x2[M] - 2*x.y ; sq = max(D + y2[N], 0)
// All min tracking is done on uint bits (valid: sq >= 0), avoiding IEEE
// minNum canonicalization VALU ops.
// grid = (NV/COLS_PER_WG, NV/(16*NWAVES), NB), block = 256.
// ---------------------------------------------------------------------------
__global__ __launch_bounds__(256) void hausdorff_main(
    const float* __restrict__ x, const float* __restrict__ y,
    unsigned* __restrict__ colminU, unsigned* __restrict__ rowminU) {

    __shared__ unsigned ldsColmin[COLS_PER_WG];

    const int tid  = threadIdx.x;
    const int wave = tid >> 5;
    const int lane = tid & 31;
    const int l16  = lane & 15;
    const int half = lane >> 4;          // 0: K=0..1 / M=0..7 ; 1: K=2..3 / M=8..15
    const int n    = blockIdx.z;
    const int rowbase = (blockIdx.y * NWAVES + wave) * 16;
    const int colbase = blockIdx.x * COLS_PER_WG;

    for (int i = tid; i < COLS_PER_WG; i += blockDim.x) ldsColmin[i] = INF_BITS;
    __syncthreads();

    // ---- A tile (branchless: 2 unconditional b32 loads per lane) -----------
    const float* xrow = x + ((size_t)n * NV + rowbase + l16) * 3;
    float xa = xrow[2 * half];           // half0: x0 ; half1: x2
    float xm = xrow[1];                  // x1 (value unused in half1)
    float xt = half ? 0.0f : xm;
    float x2p = xa * xa + xt * xt;       // half0: x0^2+x1^2 ; half1: x2^2
    float x2f = x2p + SWZF(x2p, SWZ_XOR16);   // full |x_row|^2, both halves

    v2f a;
    a.x = -2.0f * xa;
    a.y = half ? x2f : (-2.0f * xm);

    unsigned regmin[8];
    #pragma unroll
    for (int r = 0; r < 8; ++r) regmin[r] = INF_BITS;

    // per-lane y pointers; per-tile stride = 16 cols * 3 floats = 48
    const float* ypA = y + ((size_t)n * NV + colbase + l16) * 3 + 2 * half;
    const float* ypM = y + ((size_t)n * NV + colbase + l16) * 3 + 1;

    #pragma unroll 4
    for (int jt = 0; jt < NTILES; ++jt) {
        float ya = ypA[jt * 48];         // half0: y0 ; half1: y2
        float ym = ypM[jt * 48];         // y1
        float yt = half ? 0.0f : ym;
        float y2p = ya * ya + yt * yt;
        float y2  = y2p + SWZF(y2p, SWZ_XOR16);   // full |y_col|^2

        v2f b;
        b.x = ya;
        b.y = half ? 1.0f : ym;

        v8f c = {};
        v8f d = __builtin_amdgcn_wmma_f32_16x16x4_f32(
            /*neg_a=*/false, a, /*neg_b=*/false, b,
            /*c_mod=*/(short)0, c, /*reuse_a=*/false, /*reuse_b=*/false);

        unsigned cmin = INF_BITS;
        #pragma unroll
        for (int r = 0; r < 8; ++r) {
            unsigned squ = __float_as_uint(fmaxf(d[r] + y2, 0.0f));
            regmin[r] = umin2(regmin[r], squ);   // v_min_u32: no canonicalize
            cmin = umin2(cmin, squ);
        }
        // combine the two row-halves; both halves then hold the same value,
        // so all 32 lanes hit the LDS atomic (no exec-mask branch)
        cmin = umin2(cmin, SWZU(cmin, SWZ_XOR16));
        atomicMin(&ldsColmin[jt * 16 + l16], cmin);
    }

    // ---- row mins: swizzle xor-tree within each 16-lane half (uint min) ----
    #pragma unroll
    for (int r = 0; r < 8; ++r) {
        unsigned m = regmin[r];
        m = umin2(m, SWZU(m, SWZ_XOR1));
        m = umin2(m, SWZU(m, SWZ_XOR2));
        m = umin2(m, SWZU(m, SWZ_XOR4));
        m = umin2(m, SWZU(m, SWZ_XOR8));
        if (l16 == 0)
            atomicMin(&rowminU[(size_t)n * NV + rowbase + 8 * half + r], m);
    }

    // ---- flush WG-combined column mins to global (1 atomic per col per WG)
    __syncthreads();
    for (int i = tid; i < COLS_PER_WG; i += blockDim.x)
        atomicMin(&colminU[(size_t)n * NV + colbase + i], ldsColmin[i]);
}

// ---------------------------------------------------------------------------
// Final reduce: loss_n = sqrt(max over colmin[n] ++ rowmin[n]); out = mean_n.
// ---------------------------------------------------------------------------
__global__ __launch_bounds__(256) void hausdorff_reduce(
    const unsigned* __restrict__ colminU, const unsigned* __restrict__ rowminU,
    float* __restrict__ out) {

    __shared__ unsigned sred[256];
    const int tid = threadIdx.x;
    float sum = 0.0f;

    for (int n = 0; n < NB; ++n) {
        unsigned m = 0u;   // all stored values >= 0.0f, so 0 is the identity
        const unsigned* c = colminU + (size_t)n * NV;
        const unsigned* r = rowminU + (size_t)n * NV;
        for (int i = tid; i < NV; i += 256) {
            unsigned a = c[i]; if (a > m) m = a;
            unsigned b = r[i]; if (b > m) m = b;
        }
        sred[tid] = m;
        __syncthreads();
        for (int s = 128; s > 0; s >>= 1) {
            if (tid < s) { unsigned o = sred[tid + s]; if (o > sred[tid]) sred[tid] = o; }
            __syncthreads();
        }
        if (tid == 0) sum += sqrtf(__uint_as_float(sred[0]));
        __syncthreads();
    }
    if (tid == 0) *out = sum * (1.0f / (float)NB);   // LOSS_WEIGHT == 1.0
}

// ---------------------------------------------------------------------------
extern "C" void kernel_launch(void* const* d_in, const int* in_sizes, int n_in,
                              void* d_out, int out_size, void* d_ws, size_t ws_size,
                              hipStream_t stream) {
    const float* x = (const float*)d_in[0];
    const float* y = (const float*)d_in[1];
    float* out = (float*)d_out;

    unsigned* colminU = (unsigned*)d_ws;               // NB*NV uints
    unsigned* rowminU = colminU + (size_t)NB * NV;     // NB*NV uints (256 KB total)

    const int initCount = 2 * NB * NV;
    hausdorff_init<<<(initCount + 255) / 256, 256, 0, stream>>>(colminU, initCount);

    dim3 grid(NV / COLS_PER_WG, NV / (16 * NWAVES), NB);
    hausdorff_main<<<grid, 256, 0, stream>>>(x, y, colminU, rowminU);

    hausdorff_reduce<<<1, 256, 0, stream>>>(colminU, rowminU, out);
}